// EdgeMLP_44427141709913
// MI455X (gfx1250) — compile-verified
//
#include <hip/hip_runtime.h>

// EdgeMLP: out[i,j] = sigmoid( sum_k relu(pi[i,k] + pj[j,k]) * W2[k] + b2 )
// where pi = z @ W1[:64] + b1, pj = z @ W1[64:].
// N = 2048, Z_DIM = 64, HID = 64, all fp32.

#define NN  2048
#define ZD  64
#define HID 64

typedef float v2f __attribute__((ext_vector_type(2)));
typedef float v8f __attribute__((ext_vector_type(8)));

// ---------------------------------------------------------------------------
// Kernel 1: P = z @ W1 (fp32 WMMA 16x16x4, exact fp32 math).
// blockIdx.y == 0 -> Pi (W1 rows 0..63, + b1), == 1 -> Pj (W1 rows 64..127).
// One wave per 16(M) x 16(N) tile; 4 waves/block cover all 64 output columns.
// ---------------------------------------------------------------------------
__global__ __launch_bounds__(128)
void edge_proj_wmma(const float* __restrict__ z,
                    const float* __restrict__ W1,
                    const float* __restrict__ b1,
                    float* __restrict__ Pi,
                    float* __restrict__ Pj)
{
    const int lane = threadIdx.x & 31;
    const int wave = threadIdx.x >> 5;
    const int half = lane >> 4;     // 0: K pair {0,1}; 1: K pair {2,3}
    const int l15  = lane & 15;     // A: row M; B/C/D: column N
    const int i0   = blockIdx.x * 16;
    const int n0   = wave * 16;
    const int which = blockIdx.y;
    const int koff  = which * ZD;   // row offset into W1 (128 x 64 row-major)

    const float* zrow = z + (i0 + l15) * ZD;

    v8f acc = {};
    #pragma unroll
    for (int kk = 0; kk < ZD; kk += 4) {
        const int ka = kk + 2 * half;
        v2f a, b;
        // A fragment (16x4 f32): lanes 0-15 hold K={0,1}, lanes 16-31 K={2,3}
        a.x = zrow[ka + 0];
        a.y = zrow[ka + 1];
        // B fragment (4x16 f32): mirrored layout, lane = column
        b.x = W1[(koff + ka + 0) * HID + n0 + l15];
        b.y = W1[(koff + ka + 1) * HID + n0 + l15];
        acc = __builtin_amdgcn_wmma_f32_16x16x4_f32(
                  /*neg_a=*/false, a, /*neg_b=*/false, b,
                  /*c_mod=*/(short)0, acc, /*reuse_a=*/false, /*reuse_b=*/false);
    }

    // C/D layout: lane holds one column; VGPR r is row (r + 8*half).
    const float bias = (which == 0) ? b1[n0 + l15] : 0.0f;
    float* __restrict__ dst = (which == 0) ? Pi : Pj;
    #pragma unroll
    for (int r = 0; r < 8; ++r) {
        const int row = i0 + r + 8 * half;
        dst[row * HID + n0 + l15] = acc[r] + bias;
    }
}

// ---------------------------------------------------------------------------
// Kernel 2: epilogue. 64x64 output tile per 256-thread block; each thread
// owns a 4x4 register tile. pi/pj tiles live in LDS (row stride 68 floats =
// 272B to break the 64-bank * 4B = 256B period for the pj column reads).
// ---------------------------------------------------------------------------
#define LSTR 68

__global__ __launch_bounds__(256)
void edge_tile(const float* __restrict__ Pi,
               const float* __restrict__ Pj,
               const float* __restrict__ W2,
               const float* __restrict__ b2,
               float* __restrict__ out)
{
    __shared__ float sPi[64 * LSTR];
    __shared__ float sPj[64 * LSTR];
    __shared__ float sW[HID];

    const int i0 = blockIdx.y * 64;
    const int j0 = blockIdx.x * 64;

    // Stage tiles: 64 rows x 16 float4 per tile, 1024 chunks over 256 threads.
    for (int c = threadIdx.x; c < 64 * 16; c += 256) {
        const int row = c >> 4;
        const int k4  = (c & 15) << 2;
        const float4 vi = *(const float4*)(Pi + (i0 + row) * HID + k4);
        const float4 vj = *(const float4*)(Pj + (j0 + row) * HID + k4);
        *(float4*)(sPi + row * LSTR + k4) = vi;
        *(float4*)(sPj + row * LSTR + k4) = vj;
    }
    if (threadIdx.x < HID / 4) {
        *(float4*)(sW + threadIdx.x * 4) = *(const float4*)(W2 + threadIdx.x * 4);
    }
    __syncthreads();

    const int ti = (threadIdx.x >> 4) << 2;   // 16 row-groups of 4
    const int tj = (threadIdx.x & 15) << 2;   // 16 col-groups of 4

    float acc[4][4];
    #pragma unroll
    for (int r = 0; r < 4; ++r)
        #pragma unroll
        for (int c = 0; c < 4; ++c) acc[r][c] = 0.0f;

    #pragma unroll 4
    for (int kk = 0; kk < HID; kk += 4) {
        const float4 w = *(const float4*)(sW + kk);
        float4 a0 = *(const float4*)(sPi + (ti + 0) * LSTR + kk);
        float4 a1 = *(const float4*)(sPi + (ti + 1) * LSTR + kk);
        float4 a2 = *(const float4*)(sPi + (ti + 2) * LSTR + kk);
        float4 a3 = *(const float4*)(sPi + (ti + 3) * LSTR + kk);
        float4 b0 = *(const float4*)(sPj + (tj + 0) * LSTR + kk);
        float4 b1v = *(const float4*)(sPj + (tj + 1) * LSTR + kk);
        float4 b2v = *(const float4*)(sPj + (tj + 2) * LSTR + kk);
        float4 b3v = *(const float4*)(sPj + (tj + 3) * LSTR + kk);

        const float4 ar[4] = {a0, a1, a2, a3};
        const float4 bc[4] = {b0, b1v, b2v, b3v};
        #pragma unroll
        for (int r = 0; r < 4; ++r) {
            #pragma unroll
            for (int c = 0; c < 4; ++c) {
                acc[r][c] = fmaf(fmaxf(ar[r].x + bc[c].x, 0.0f), w.x, acc[r][c]);
                acc[r][c] = fmaf(fmaxf(ar[r].y + bc[c].y, 0.0f), w.y, acc[r][c]);
                acc[r][c] = fmaf(fmaxf(ar[r].z + bc[c].z, 0.0f), w.z, acc[r][c]);
                acc[r][c] = fmaf(fmaxf(ar[r].w + bc[c].w, 0.0f), w.w, acc[r][c]);
            }
        }
    }

    const float bb = b2[0];
    #pragma unroll
    for (int r = 0; r < 4; ++r) {
        float4 o;
        o.x = 1.0f / (1.0f + __expf(-(acc[r][0] + bb)));
        o.y = 1.0f / (1.0f + __expf(-(acc[r][1] + bb)));
        o.z = 1.0f / (1.0f + __expf(-(acc[r][2] + bb)));
        o.w = 1.0f / (1.0f + __expf(-(acc[r][3] + bb)));
        *(float4*)(out + (size_t)(i0 + ti + r) * NN + j0 + tj) = o;
    }
}

// ---------------------------------------------------------------------------
extern "C" void kernel_launch(void* const* d_in, const int* in_sizes, int n_in,
                              void* d_out, int out_size, void* d_ws, size_t ws_size,
                              hipStream_t stream)
{
    (void)in_sizes; (void)n_in; (void)out_size; (void)ws_size;

    const float* z  = (const float*)d_in[0];   // (2048, 64)
    const float* W1 = (const float*)d_in[1];   // (128, 64)
    const float* b1 = (const float*)d_in[2];   // (64,)
    const float* W2 = (const float*)d_in[3];   // (64, 1)
    const float* b2 = (const float*)d_in[4];   // (1,)
    float* out = (float*)d_out;                // (2048, 2048)

    float* Pi = (float*)d_ws;                  // (2048, 64) fp32
    float* Pj = Pi + (size_t)NN * HID;         // (2048, 64) fp32

    // Kernel 1: WMMA projections (Pi gets b1 folded in).
    edge_proj_wmma<<<dim3(NN / 16, 2), 128, 0, stream>>>(z, W1, b1, Pi, Pj);

    // Kernel 2: fused broadcast-add + relu + dot(W2) + sigmoid epilogue.
    edge_tile<<<dim3(NN / 64, NN / 64), 256, 0, stream>>>(Pi, Pj, W2, b2, out);
}